// CMCMem_87058987090687
// MI455X (gfx1250) — compile-verified
//
#include <hip/hip_runtime.h>

typedef __attribute__((ext_vector_type(2))) float v2f;
typedef __attribute__((ext_vector_type(4))) float v4f;
typedef __attribute__((ext_vector_type(8))) float v8f;

#define BSZ 32
#define NDIM 128
#define KMAX 65536          // K
#define KP1 65537           // K+1 slots per batch row
#define NTILES 4097         // ceil(65537/16)
#define INV_T 14.285714285714285714f   // 1/0.07

static constexpr long long OFF_L1  = 0;                          // logits1 [32,65537]
static constexpr long long OFF_L2  = (long long)BSZ * KP1;       // logits2
static constexpr long long OFF_LB  = 2LL * BSZ * KP1;            // labels [32]
static constexpr long long OFF_NM1 = OFF_LB + BSZ;               // new_memory_1 [1e6,128]
static constexpr long long MEMN    = 128000000LL;                // 1e6 * 128
static constexpr long long OFF_NM2 = OFF_NM1 + MEMN;             // new_memory_2

// ---------------------------------------------------------------------------
// Kernel 1: gathered dot-product logits via V_WMMA_F32_16X16X4_F32.
// One wave per (bank, batch row b, tile of 16 k-slots).
//   bank 0: logits1 = (memory_2[idx] . x1) / T
//   bank 1: logits2 = (memory_1[idx] . x2) / T
// A: 16 gathered rows (lane m / m+16 hold row m, K split {0,1}/{2,3})
// B: x[b] chunk broadcast into every column
// D: all columns identical partial dots; lane0 carries slots 0-7, lane16 8-15.
// ---------------------------------------------------------------------------
__global__ __launch_bounds__(256) void cmc_logits_wmma(
    const float* __restrict__ x1, const float* __restrict__ x2,
    const int* __restrict__ y,   const int* __restrict__ idx,
    const float* __restrict__ mem1, const float* __restrict__ mem2,
    float* __restrict__ out)
{
    const int lane = threadIdx.x & 31;
    const int warp = threadIdx.x >> 5;
    const int unit = blockIdx.x * 8 + warp;

    const int UNITS_PER_BANK = BSZ * NTILES;      // 131,104
    const int bank = (unit >= UNITS_PER_BANK) ? 1 : 0;
    const int u    = unit - bank * UNITS_PER_BANK;
    const int b    = u / NTILES;
    const int tile = u - b * NTILES;

    const float* __restrict__ mem = bank ? mem1 : mem2; // logits1 gathers memory_2
    const float* __restrict__ x   = bank ? x2   : x1;
    float* __restrict__ obase     = out + (bank ? OFF_L2 : OFF_L1);

    const int m  = lane & 15;     // slot within tile / A-matrix row
    const int hi = lane >> 4;     // K-half select

    // slot index; column 0 is replaced by the positive index y[b]
    const int k  = tile * 16 + m;
    const int kc = (k <= KMAX) ? k : 0;
    int g = idx[(long long)b * KP1 + kc];
    g = (k <= KMAX) ? g : 0;
    g = (k == 0) ? y[b] : g;

    const float* __restrict__ wrow = mem + (long long)g * NDIM + 2 * hi;
    const float* __restrict__ xrow = x + b * NDIM + 2 * hi;

    v8f c0 = {}; v8f c1 = {}; v8f c2 = {}; v8f c3 = {};
    #pragma unroll
    for (int t = 0; t < 32; t += 4) {
        v2f a0 = *(const v2f*)(wrow + 4 * (t + 0));
        v2f b0 = *(const v2f*)(xrow + 4 * (t + 0));
        c0 = __builtin_amdgcn_wmma_f32_16x16x4_f32(false, a0, false, b0,
                                                   (short)0, c0, false, false);
        v2f a1 = *(const v2f*)(wrow + 4 * (t + 1));
        v2f b1 = *(const v2f*)(xrow + 4 * (t + 1));
        c1 = __builtin_amdgcn_wmma_f32_16x16x4_f32(false, a1, false, b1,
                                                   (short)0, c1, false, false);
        v2f a2 = *(const v2f*)(wrow + 4 * (t + 2));
        v2f b2 = *(const v2f*)(xrow + 4 * (t + 2));
        c2 = __builtin_amdgcn_wmma_f32_16x16x4_f32(false, a2, false, b2,
                                                   (short)0, c2, false, false);
        v2f a3 = *(const v2f*)(wrow + 4 * (t + 3));
        v2f b3 = *(const v2f*)(xrow + 4 * (t + 3));
        c3 = __builtin_amdgcn_wmma_f32_16x16x4_f32(false, a3, false, b3,
                                                   (short)0, c3, false, false);
    }
    v8f c = (c0 + c1) + (c2 + c3);

    // D layout: lanes 0-15 -> M = r (VGPR r); lanes 16-31 -> M = r + 8.
    // All columns are identical, so lane 0 and lane 16 store the 16 logits.
    if (m == 0) {
        const int base_slot = tile * 16 + hi * 8;
        float* o = obase + (long long)b * KP1 + base_slot;
        #pragma unroll
        for (int r = 0; r < 8; ++r) {
            if (base_slot + r <= KMAX) o[r] = c[r] * INV_T;
        }
    }
}

// ---------------------------------------------------------------------------
// Kernel 2: stream both memory banks into the output (non-temporal so the
// 2 GB stream doesn't evict the gather working set from L2), zero the labels.
// ---------------------------------------------------------------------------
__global__ __launch_bounds__(256) void cmc_copy(
    const float* __restrict__ mem1, const float* __restrict__ mem2,
    float* __restrict__ out)
{
    const long long half = MEMN / 4;               // 32,000,000 float4 per bank
    const v4f* __restrict__ s1 = (const v4f*)mem1;
    const v4f* __restrict__ s2 = (const v4f*)mem2;
    v4f* __restrict__ d1 = (v4f*)(out + OFF_NM1);
    v4f* __restrict__ d2 = (v4f*)(out + OFF_NM2);

    const long long i0     = (long long)blockIdx.x * blockDim.x + threadIdx.x;
    const long long stride = (long long)gridDim.x * blockDim.x;
    for (long long i = i0; i < half; i += stride) {
        v4f a = __builtin_nontemporal_load(s1 + i);
        __builtin_nontemporal_store(a, d1 + i);
        v4f b = __builtin_nontemporal_load(s2 + i);
        __builtin_nontemporal_store(b, d2 + i);
    }
    if (i0 < BSZ) out[OFF_LB + i0] = 0.0f;         // labels = zeros
}

// ---------------------------------------------------------------------------
// Kernel 3: EMA update of the 2x32 touched rows, L2-normalize, scatter into
// the already-copied output banks. One wave per row; wave32 shfl reduction.
// ---------------------------------------------------------------------------
__global__ __launch_bounds__(256) void cmc_update(
    const float* __restrict__ x1, const float* __restrict__ x2,
    const int* __restrict__ y,
    const float* __restrict__ mem1, const float* __restrict__ mem2,
    float* __restrict__ out)
{
    const int lane = threadIdx.x & 31;
    const int wave = (blockIdx.x * blockDim.x + threadIdx.x) >> 5; // 0..63
    const int bank = wave >> 5;
    const int b    = wave & 31;

    const float* __restrict__ mem = bank ? mem2 : mem1;
    const float* __restrict__ x   = bank ? x2   : x1;
    float* __restrict__ o         = out + (bank ? OFF_NM2 : OFF_NM1);

    const int row = y[b];
    const float* __restrict__ src = mem + (long long)row * NDIM + lane * 4;
    const float* __restrict__ xs  = x + b * NDIM + lane * 4;

    float w0 = 0.5f * (src[0] + xs[0]);
    float w1 = 0.5f * (src[1] + xs[1]);
    float w2 = 0.5f * (src[2] + xs[2]);
    float w3 = 0.5f * (src[3] + xs[3]);

    float s = w0 * w0 + w1 * w1 + w2 * w2 + w3 * w3;
    #pragma unroll
    for (int off = 16; off > 0; off >>= 1) s += __shfl_xor(s, off, 32);
    const float inv = 1.0f / sqrtf(s);

    v4f r = {w0 * inv, w1 * inv, w2 * inv, w3 * inv};
    *(v4f*)(o + (long long)row * NDIM + lane * 4) = r;
}

// ---------------------------------------------------------------------------
extern "C" void kernel_launch(void* const* d_in, const int* in_sizes, int n_in,
                              void* d_out, int out_size, void* d_ws, size_t ws_size,
                              hipStream_t stream) {
    const float* x1  = (const float*)d_in[0];
    const float* x2  = (const float*)d_in[1];
    const int*   y   = (const int*)d_in[2];
    const int*   idx = (const int*)d_in[3];
    const float* m1  = (const float*)d_in[4];
    const float* m2  = (const float*)d_in[5];
    float* out = (float*)d_out;

    // 2 banks * 32 batch rows * 4097 tiles = 262,208 waves; 8 waves / block
    const int units  = 2 * BSZ * NTILES;
    const int blocks = units / 8;                   // 32,776 (exact)
    hipLaunchKernelGGL(cmc_logits_wmma, dim3(blocks), dim3(256), 0, stream,
                       x1, x2, y, idx, m1, m2, out);

    hipLaunchKernelGGL(cmc_copy, dim3(8192), dim3(256), 0, stream, m1, m2, out);

    // 64 waves = 2048 threads; must follow the copy (same-stream ordering)
    hipLaunchKernelGGL(cmc_update, dim3(8), dim3(256), 0, stream,
                       x1, x2, y, m1, m2, out);
}